// GraphSAGECustom_47648367182185
// MI455X (gfx1250) — compile-verified
//
#include <hip/hip_runtime.h>
#include <hip/hip_bf16.h>

#define N_NODES 100000
#define N_EDGES 600000
#define D_IN    128
#define D_HID   128
#define D_OUT   16

typedef __attribute__((ext_vector_type(2))) float v2f;
typedef __attribute__((ext_vector_type(8))) float v8f;

// ---------------------------------------------------------------------------
// Degree count: one f32 atomic per edge (lands in L2 atomic units).
// ---------------------------------------------------------------------------
__global__ void count_deg_kernel(const int* __restrict__ dst,
                                 float* __restrict__ cnt, int E) {
    int e = blockIdx.x * blockDim.x + threadIdx.x;
    if (e < E) atomicAdd(&cnt[dst[e]], 1.0f);
}

__global__ void inv_cnt_kernel(float* __restrict__ cnt, int n) {
    int i = blockIdx.x * blockDim.x + threadIdx.x;
    if (i < n) cnt[i] = 1.0f / fmaxf(cnt[i], 1.0f);
}

// ---------------------------------------------------------------------------
// Scatter-add of messages: one wave32 per edge, 32 lanes x float4 = 128 feats.
// b128 gather loads + global_atomic_add_f32 scatter; both hot in 192MB L2.
// ---------------------------------------------------------------------------
__global__ __launch_bounds__(256)
void scatter_add_kernel(const float* __restrict__ feat,
                        const int* __restrict__ src,
                        const int* __restrict__ dst,
                        float* __restrict__ agg, int E) {
    int e    = blockIdx.x * 8 + (threadIdx.x >> 5);
    int lane = threadIdx.x & 31;
    if (e >= E) return;
    int s = src[e], d = dst[e];
    const float4 m = ((const float4*)(feat + (size_t)s * D_IN))[lane];
    float* ap = agg + (size_t)d * D_IN + lane * 4;
    atomicAdd(ap + 0, m.x);
    atomicAdd(ap + 1, m.y);
    atomicAdd(ap + 2, m.z);
    atomicAdd(ap + 3, m.w);
}

// ---------------------------------------------------------------------------
// Fused SAGE layer: out = act( (agg*inv) @ Wl^T + b + x @ Wr^T )
// One wave per 16x16 output tile; f32 WMMA (16x16x4), K=128 -> 64 WMMAs/tile.
// A frag (16x4 f32): lane<16 holds K=kb..kb+1, lane>=16 K=kb+2..kb+3 (float2).
// B frag (4x16 f32): column j = lane&15, same K split -> contiguous float2 of
//   W[j][kb+off .. kb+off+1] since B = W^T.
// C/D frag: 8 VGPRs, M = v + 8*(lane>=16), Ncol = j0 + (lane&15).
// ---------------------------------------------------------------------------
template<int DOUT, bool RELU>
__global__ __launch_bounds__(256)
void sage_layer_kernel(const float* __restrict__ agg,
                       const float* __restrict__ xin,
                       const float* __restrict__ Wl,
                       const float* __restrict__ bias,
                       const float* __restrict__ Wr,
                       const float* __restrict__ inv,
                       float* __restrict__ out) {
    constexpr int COL_TILES = DOUT / 16;
    constexpr int ROW_TILES = N_NODES / 16;      // 100000/16 = 6250 exact
    const int wave = threadIdx.x >> 5;           // wave-uniform
    const int lane = threadIdx.x & 31;
    const int tile = blockIdx.x * 8 + wave;
    if (tile >= ROW_TILES * COL_TILES) return;   // wave-uniform exit: EXEC all-1s inside
    const int rt = tile / COL_TILES;
    const int ct = tile - rt * COL_TILES;
    const int n0 = rt << 4;
    const int j0 = ct << 4;

    const int r    = lane & 15;                  // A row within tile / B column j
    const int koff = (lane >> 4) << 1;           // 0 or 2: K sub-offset per lane half
    const float invc = inv[n0 + r];              // mean = agg * (1/deg), folded into A

    const float* aggp = agg + (size_t)(n0 + r) * D_IN + koff;
    const float* xp   = xin + (size_t)(n0 + r) * D_IN + koff;
    const float* wlp  = Wl  + (size_t)(j0 + r) * D_IN + koff;
    const float* wrp  = Wr  + (size_t)(j0 + r) * D_IN + koff;

    v8f c = {};
#pragma unroll 4
    for (int kb = 0; kb < D_IN; kb += 4) {
        v2f al = *(const v2f*)(aggp + kb);
        al.x *= invc; al.y *= invc;
        v2f ar = *(const v2f*)(xp + kb);
        v2f bl = *(const v2f*)(wlp + kb);
        v2f br = *(const v2f*)(wrp + kb);
        c = __builtin_amdgcn_wmma_f32_16x16x4_f32(false, al, false, bl,
                                                  (short)0, c, false, false);
        c = __builtin_amdgcn_wmma_f32_16x16x4_f32(false, ar, false, br,
                                                  (short)0, c, false, false);
    }

    const float b  = bias[j0 + r];
    const int mbase = n0 + ((lane >> 4) << 3);   // +8 rows for upper lane half
#pragma unroll
    for (int v = 0; v < 8; ++v) {
        float val = c[v] + b;
        if (RELU) val = fmaxf(val, 0.0f);
        out[(size_t)(mbase + v) * DOUT + j0 + r] = val;
    }
}

// ---------------------------------------------------------------------------
// In-place log_softmax over 16-wide rows.
// ---------------------------------------------------------------------------
__global__ void log_softmax_kernel(float* __restrict__ out, int n) {
    int i = blockIdx.x * blockDim.x + threadIdx.x;
    if (i >= n) return;
    float* row = out + (size_t)i * D_OUT;
    float v[D_OUT];
    float m = -3.4e38f;
#pragma unroll
    for (int j = 0; j < D_OUT; ++j) { v[j] = row[j]; m = fmaxf(m, v[j]); }
    float s = 0.0f;
#pragma unroll
    for (int j = 0; j < D_OUT; ++j) s += __expf(v[j] - m);
    const float lse = m + __logf(s);
#pragma unroll
    for (int j = 0; j < D_OUT; ++j) row[j] = v[j] - lse;
}

// ---------------------------------------------------------------------------
extern "C" void kernel_launch(void* const* d_in, const int* in_sizes, int n_in,
                              void* d_out, int out_size, void* d_ws, size_t ws_size,
                              hipStream_t stream) {
    (void)in_sizes; (void)n_in; (void)out_size; (void)ws_size;

    const float* x   = (const float*)d_in[0];
    const int*   ei  = (const int*)  d_in[1];   // [2, E]
    const float* Wl0 = (const float*)d_in[2];
    const float* bl0 = (const float*)d_in[3];
    const float* Wr0 = (const float*)d_in[4];
    const float* Wl1 = (const float*)d_in[5];
    const float* bl1 = (const float*)d_in[6];
    const float* Wr1 = (const float*)d_in[7];
    const float* Wl2 = (const float*)d_in[8];
    const float* bl2 = (const float*)d_in[9];
    const float* Wr2 = (const float*)d_in[10];
    const int* src = ei;
    const int* dst = ei + N_EDGES;

    // Workspace layout (floats): inv[N] | agg[N*128] | h1[N*128] | h2[N*128]
    float* ws  = (float*)d_ws;
    float* inv = ws;
    float* agg = ws  + N_NODES;
    float* h1  = agg + (size_t)N_NODES * D_IN;
    float* h2  = h1  + (size_t)N_NODES * D_IN;
    float* logits = (float*)d_out;

    const size_t aggBytes = (size_t)N_NODES * D_IN * sizeof(float);
    const int cblk = (N_EDGES + 255) / 256;
    const int nblk = (N_NODES + 255) / 256;
    const int sblk = (N_EDGES + 7) / 8;                 // 8 edges per 256-thr block
    const int gblk_hid = (N_NODES / 16) * (D_HID / 16) / 8;       // 6250
    const int gblk_out = ((N_NODES / 16) * (D_OUT / 16) + 7) / 8; // 782

    // degrees -> 1/max(deg,1)   (computed once; edges identical across layers)
    hipMemsetAsync(inv, 0, N_NODES * sizeof(float), stream);
    count_deg_kernel<<<cblk, 256, 0, stream>>>(dst, inv, N_EDGES);
    inv_cnt_kernel<<<nblk, 256, 0, stream>>>(inv, N_NODES);

    // layer 0
    hipMemsetAsync(agg, 0, aggBytes, stream);
    scatter_add_kernel<<<sblk, 256, 0, stream>>>(x, src, dst, agg, N_EDGES);
    sage_layer_kernel<D_HID, true><<<gblk_hid, 256, 0, stream>>>(agg, x, Wl0, bl0, Wr0, inv, h1);

    // layer 1
    hipMemsetAsync(agg, 0, aggBytes, stream);
    scatter_add_kernel<<<sblk, 256, 0, stream>>>(h1, src, dst, agg, N_EDGES);
    sage_layer_kernel<D_HID, true><<<gblk_hid, 256, 0, stream>>>(agg, h1, Wl1, bl1, Wr1, inv, h2);

    // layer 2 + log_softmax
    hipMemsetAsync(agg, 0, aggBytes, stream);
    scatter_add_kernel<<<sblk, 256, 0, stream>>>(h2, src, dst, agg, N_EDGES);
    sage_layer_kernel<D_OUT, false><<<gblk_out, 256, 0, stream>>>(agg, h2, Wl2, bl2, Wr2, inv, logits);
    log_softmax_kernel<<<nblk, 256, 0, stream>>>(logits, N_NODES);
}